// PatchExtractor_19481971654740
// MI455X (gfx1250) — compile-verified
//
#include <hip/hip_runtime.h>
#include <stdint.h>

#define IMG_W 2048
#define IMG_H 2048
#define OUT_S 224
#define TILE  32
#define NT    7            // 224 / 32
#define MAX_FOOT 48        // footprint per 32x32 tile: <= 47 (scale <= 319/224)

// ---- CDNA5 async global->LDS copy (ASYNCcnt-tracked) ------------------------
__device__ __forceinline__ void async_g2l_b32(const float* gsrc, float* ldst) {
#if __has_builtin(__builtin_amdgcn_global_load_async_to_lds_b32)
  typedef __attribute__((address_space(1))) int gint;
  typedef __attribute__((address_space(3))) int lint;
  __builtin_amdgcn_global_load_async_to_lds_b32(
      (gint*)(uintptr_t)gsrc,
      (lint*)(uint32_t)(uintptr_t)ldst, 0, 0);
#else
  uint32_t loff = (uint32_t)(uintptr_t)ldst;   // low 32 bits of generic = LDS byte offset
  asm volatile("global_load_async_to_lds_b32 %0, %1, off"
               :: "v"(loff), "v"(gsrc) : "memory");
#endif
}

__device__ __forceinline__ void wait_async0() {
#if __has_builtin(__builtin_amdgcn_s_wait_asynccnt)
  __builtin_amdgcn_s_wait_asynccnt(0);
#else
  asm volatile("s_wait_asynccnt 0" ::: "memory");
#endif
}

// ---- kernel -----------------------------------------------------------------
__global__ __launch_bounds__(256)
void patch_extract_kernel(const float* __restrict__ img,
                          const float* __restrict__ mean,
                          const float* __restrict__ stdv,
                          const int*   __restrict__ bboxes,
                          float* __restrict__ out)
{
  __shared__ float lds[MAX_FOOT * MAX_FOOT];

  const int tid = threadIdx.x;
  const int tIdx = blockIdx.x;          // 0..48 : 7x7 output tiles
  const int c    = blockIdx.y;          // channel
  const int n    = blockIdx.z;          // patch
  const int ty = tIdx / NT;
  const int tx = tIdx - ty * NT;

  const int4 bb = ((const int4*)bboxes)[n];
  const float x0 = (float)bb.x, y0 = (float)bb.y;
  const float x1 = (float)bb.z, y1 = (float)bb.w;
  const float side  = fminf(x1 - x0, y1 - y0);
  const float scale = side * (1.0f / (float)OUT_S);
  const float bx = 0.5f * (x0 + x1) - 0.5f * side - 0.5f;   // sx(j) = bx + (j+0.5)*scale
  const float by = 0.5f * (y0 + y1) - 0.5f * side - 0.5f;

  const int j0 = tx * TILE;
  const int i0 = ty * TILE;

  // Staged source rectangle (always fully inside the image after clipping).
  const float sxf = fminf(fmaxf(bx + ((float)j0 + 0.5f) * scale, 0.0f), (float)(IMG_W - 1));
  const float sxl = fminf(fmaxf(bx + ((float)j0 + TILE - 0.5f) * scale, 0.0f), (float)(IMG_W - 1));
  const float syf = fminf(fmaxf(by + ((float)i0 + 0.5f) * scale, 0.0f), (float)(IMG_H - 1));
  const float syl = fminf(fmaxf(by + ((float)i0 + TILE - 0.5f) * scale, 0.0f), (float)(IMG_H - 1));
  const int xs0 = (int)floorf(sxf);
  const int xs1 = min((int)floorf(sxl) + 1, IMG_W - 1);
  const int ys0 = (int)floorf(syf);
  const int ys1 = min((int)floorf(syl) + 1, IMG_H - 1);
  const int ncols = xs1 - xs0 + 1;      // <= 47
  const int nrows = ys1 - ys0 + 1;      // <= 47
  const int total = nrows * ncols;

  const float* gbase = img + (size_t)c * (IMG_W * IMG_H)
                           + (size_t)ys0 * IMG_W + xs0;

  // Async-stage the footprint into LDS (row-contiguous global reads).
  for (int idx = tid; idx < total; idx += 256) {
    const int r  = (int)((unsigned)idx / (unsigned)ncols);
    const int cc = idx - r * ncols;
    async_g2l_b32(gbase + (size_t)r * IMG_W + cc, &lds[idx]);
  }
  wait_async0();
  __syncthreads();

  // Fused normalization: (p/255 - mu)/sigma = p*gain + bias
  const float sd   = stdv[c];
  const float gain = 1.0f / (255.0f * sd);
  const float bias = -mean[c] / sd;

  // x-interp setup once per thread (lane -> fixed output column).
  const int lc = tid & (TILE - 1);
  const int j  = j0 + lc;
  const float sx = fminf(fmaxf(bx + ((float)j + 0.5f) * scale, 0.0f), (float)(IMG_W - 1));
  const int   ix0 = (int)floorf(sx);
  const float fx  = sx - (float)ix0;
  const int lx0 = ix0 - xs0;
  const int lx1 = min(ix0 + 1, IMG_W - 1) - xs0;

  float* oplane = out + ((size_t)n * 3 + c) * (OUT_S * OUT_S);

  #pragma unroll
  for (int k = 0; k < 4; ++k) {
    const int lr = (tid >> 5) + k * 8;
    const int i  = i0 + lr;
    const float sy = fminf(fmaxf(by + ((float)i + 0.5f) * scale, 0.0f), (float)(IMG_H - 1));
    const int   iy0 = (int)floorf(sy);
    const float fy  = sy - (float)iy0;
    const int ly0 = iy0 - ys0;
    const int ly1 = min(iy0 + 1, IMG_H - 1) - ys0;

    const float p00 = lds[ly0 * ncols + lx0];
    const float p01 = lds[ly0 * ncols + lx1];
    const float p10 = lds[ly1 * ncols + lx0];
    const float p11 = lds[ly1 * ncols + lx1];
    const float top = p00 + fx * (p01 - p00);
    const float bot = p10 + fx * (p11 - p10);
    const float v   = top + fy * (bot - top);

    oplane[(size_t)i * OUT_S + j] = fmaf(v, gain, bias);
  }
}

// ---- launch -----------------------------------------------------------------
extern "C" void kernel_launch(void* const* d_in, const int* in_sizes, int n_in,
                              void* d_out, int out_size, void* d_ws, size_t ws_size,
                              hipStream_t stream) {
  const float* img    = (const float*)d_in[0];
  const float* mean   = (const float*)d_in[1];
  const float* stdv   = (const float*)d_in[2];
  const int*   bboxes = (const int*)d_in[3];
  float* out = (float*)d_out;

  const int N = in_sizes[3] / 4;        // bboxes are [N,4]
  dim3 grid(NT * NT, 3, N);
  patch_extract_kernel<<<grid, 256, 0, stream>>>(img, mean, stdv, bboxes, out);
}